// GCN_43430709297569
// MI455X (gfx1250) — compile-verified
//
#include <hip/hip_runtime.h>

typedef __attribute__((ext_vector_type(2))) float v2f;
typedef __attribute__((ext_vector_type(8))) float v8f;

#define IN_DIMC 256
#define H_DIMC  128
#define OUT_DIMC 40
#define OUT_PADC 48

// ------------------------------------------------------------------
// utility kernels
// ------------------------------------------------------------------
__global__ void zero_f32(float* __restrict__ p, int n) {
  int i = blockIdx.x * blockDim.x + threadIdx.x;
  if (i < n) p[i] = 0.0f;
}

__global__ void init_deg(float* __restrict__ deg, int n) {
  int i = blockIdx.x * blockDim.x + threadIdx.x;
  if (i < n) deg[i] = 1.0f;  // self-loop contributes 1 to every node's degree
}

__global__ void deg_edges(const int* __restrict__ dst, float* __restrict__ deg, int e) {
  int i = blockIdx.x * blockDim.x + threadIdx.x;
  if (i < e) atomicAdd(&deg[dst[i]], 1.0f);
}

__global__ void finish_dinv(float* __restrict__ deg, int n) {
  int i = blockIdx.x * blockDim.x + threadIdx.x;
  if (i < n) deg[i] = rsqrtf(fmaxf(deg[i], 1e-12f));
}

__global__ void edge_norm(const int* __restrict__ src, const int* __restrict__ dst,
                          const float* __restrict__ dinv, float* __restrict__ norm, int e) {
  int i = blockIdx.x * blockDim.x + threadIdx.x;
  if (i < e) norm[i] = dinv[src[i]] * dinv[dst[i]];
}

// ------------------------------------------------------------------
// GEMM1: H[N,128] = X[N,256] @ W1[256,128]   (fp32 WMMA 16x16x4)
// block = 256 threads = 8 waves; one 16-row tile, wave w -> cols [16w,16w+16)
// K tiled in 64-wide LDS slabs.
// ------------------------------------------------------------------
__global__ __launch_bounds__(256) void gemm1_wmma(const float* __restrict__ X,
                                                  const float* __restrict__ W,
                                                  float* __restrict__ H, int n) {
  __shared__ float sW[64][H_DIMC + 4];  // K-slab of W (padded stride vs banks)
  __shared__ float sX[16][64 + 4];      // 16-row x 64-K tile of X

  const int tid  = threadIdx.x;
  const int wave = tid >> 5;
  const int lane = tid & 31;
  const int col0 = wave * 16;
  const int m    = lane & 15;   // row within A tile / col within B,C tile
  const int kh   = lane >> 4;   // half-select (ISA 32-bit A/B/C layouts)
  const int row0 = blockIdx.x * 16;

  v8f acc = {0.f, 0.f, 0.f, 0.f, 0.f, 0.f, 0.f, 0.f};

  for (int kb = 0; kb < IN_DIMC; kb += 64) {
    __syncthreads();
    // stage W slab: 64x128 floats
    for (int i = tid; i < 64 * H_DIMC; i += 256) {
      int r = i >> 7, c = i & (H_DIMC - 1);
      sW[r][c] = W[(size_t)(kb + r) * H_DIMC + c];
    }
    // stage X tile: 16x64 floats
    for (int i = tid; i < 16 * 64; i += 256) {
      int r = i >> 6, k = i & 63;
      int row = row0 + r;
      sX[r][k] = (row < n) ? X[(size_t)row * IN_DIMC + kb + k] : 0.0f;
    }
    __syncthreads();

#pragma unroll
    for (int ks = 0; ks < 64; ks += 4) {
      v2f a, b;
      // A 16x4 f32: lane m (+16*kh): v0 = K=2kh, v1 = K=2kh+1
      a[0] = sX[m][ks + 2 * kh + 0];
      a[1] = sX[m][ks + 2 * kh + 1];
      // B 4x16 f32: lane n (+16*kh): v0 = row K=2kh, v1 = row K=2kh+1
      b[0] = sW[ks + 2 * kh + 0][col0 + m];
      b[1] = sW[ks + 2 * kh + 1][col0 + m];
      acc = __builtin_amdgcn_wmma_f32_16x16x4_f32(false, a, false, b,
                                                  (short)0, acc, false, false);
    }
  }
  // C/D 16x16 f32: VGPR r -> row r + 8*kh, col = lane&15
#pragma unroll
  for (int r = 0; r < 8; ++r) {
    int row = row0 + 8 * kh + r;
    if (row < n) H[(size_t)row * H_DIMC + col0 + m] = acc[r];
  }
}

// ------------------------------------------------------------------
// GEMM2: H2[N,40] = Xh[N,128] @ W2[128,40]  (cols padded to 48)
// block = 384 threads = 12 waves = 4 row-tiles x 3 col-tiles; 64 rows/block.
// W2 fully LDS-resident.
// ------------------------------------------------------------------
__global__ __launch_bounds__(384) void gemm2_wmma(const float* __restrict__ Xh,
                                                  const float* __restrict__ W2,
                                                  float* __restrict__ H2, int n) {
  __shared__ float sW[H_DIMC][OUT_PADC + 4];  // 128 x 52
  __shared__ float sX[64][H_DIMC + 4];        // 64 x 132

  const int tid  = threadIdx.x;
  const int wave = tid >> 5;      // 0..11
  const int lane = tid & 31;
  const int rt   = wave / 3;      // row tile 0..3
  const int ct   = wave % 3;      // col tile 0..2
  const int col0 = ct * 16;
  const int m    = lane & 15;
  const int kh   = lane >> 4;
  const int row0 = blockIdx.x * 64;

  for (int i = tid; i < H_DIMC * OUT_PADC; i += 384) {
    int r = i / OUT_PADC, c = i % OUT_PADC;
    sW[r][c] = (c < OUT_DIMC) ? W2[(size_t)r * OUT_DIMC + c] : 0.0f;
  }
  for (int i = tid; i < 64 * H_DIMC; i += 384) {
    int r = i >> 7, k = i & (H_DIMC - 1);
    int row = row0 + r;
    sX[r][k] = (row < n) ? Xh[(size_t)row * H_DIMC + k] : 0.0f;
  }
  __syncthreads();

  v8f acc = {0.f, 0.f, 0.f, 0.f, 0.f, 0.f, 0.f, 0.f};
#pragma unroll
  for (int ks = 0; ks < H_DIMC; ks += 4) {
    v2f a, b;
    a[0] = sX[rt * 16 + m][ks + 2 * kh + 0];
    a[1] = sX[rt * 16 + m][ks + 2 * kh + 1];
    b[0] = sW[ks + 2 * kh + 0][col0 + m];
    b[1] = sW[ks + 2 * kh + 1][col0 + m];
    acc = __builtin_amdgcn_wmma_f32_16x16x4_f32(false, a, false, b,
                                                (short)0, acc, false, false);
  }
#pragma unroll
  for (int r = 0; r < 8; ++r) {
    int row = row0 + rt * 16 + 8 * kh + r;
    int col = col0 + m;
    if (row < n && col < OUT_DIMC) H2[(size_t)row * OUT_DIMC + col] = acc[r];
  }
}

// ------------------------------------------------------------------
// Edge aggregation, 128 features: one wave32 per edge.
// Lane l: float4 load (global_load_b128) of h[src] + 4 fp32 atomics.
// h and out are L2-resident (51.2 MB each on 192 MB L2).
// ------------------------------------------------------------------
__global__ __launch_bounds__(256) void agg_edges_h(const int* __restrict__ src,
                                                   const int* __restrict__ dst,
                                                   const float* __restrict__ norm,
                                                   const float* __restrict__ h,
                                                   float* __restrict__ out, int e) {
  int gw   = (blockIdx.x * 256 + threadIdx.x) >> 5;  // global wave id == edge id
  int lane = threadIdx.x & 31;
  if (gw >= e) return;
  int s = src[gw], d = dst[gw];
  float w = norm[gw];
  const float4* hp = (const float4*)(h + (size_t)s * H_DIMC);
  float4 v = hp[lane];
  float* op = out + (size_t)d * H_DIMC + lane * 4;
  atomicAdd(op + 0, v.x * w);
  atomicAdd(op + 1, v.y * w);
  atomicAdd(op + 2, v.z * w);
  atomicAdd(op + 3, v.w * w);
}

// Edge aggregation, 40 features: one wave32 per edge.
__global__ __launch_bounds__(256) void agg_edges_o(const int* __restrict__ src,
                                                   const int* __restrict__ dst,
                                                   const float* __restrict__ norm,
                                                   const float* __restrict__ h2,
                                                   float* __restrict__ out, int e) {
  int gw   = (blockIdx.x * 256 + threadIdx.x) >> 5;
  int lane = threadIdx.x & 31;
  if (gw >= e) return;
  int s = src[gw], d = dst[gw];
  float w = norm[gw];
  const float* hp = h2 + (size_t)s * OUT_DIMC;
  float* op = out + (size_t)d * OUT_DIMC;
  for (int f = lane; f < OUT_DIMC; f += 32)
    atomicAdd(op + f, hp[f] * w);
}

// self-loop + bias + ReLU (in place on agg1), total = N*128
__global__ void relu_bias1(float* __restrict__ agg1, const float* __restrict__ h1,
                           const float* __restrict__ dinv, const float* __restrict__ b1,
                           int total) {
  int i = blockIdx.x * blockDim.x + threadIdx.x;
  if (i >= total) return;
  int node = i >> 7;          // /128
  int f    = i & 127;
  float di = dinv[node];
  float v = agg1[i] + h1[i] * di * di + b1[f];
  agg1[i] = v > 0.0f ? v : 0.0f;
}

// self-loop + bias into final output (out already holds edge aggregation), total = N*40
__global__ void final_out(float* __restrict__ out, const float* __restrict__ h2,
                          const float* __restrict__ dinv, const float* __restrict__ b2,
                          int total) {
  int i = blockIdx.x * blockDim.x + threadIdx.x;
  if (i >= total) return;
  int node = i / OUT_DIMC;
  int f    = i - node * OUT_DIMC;
  float di = dinv[node];
  out[i] += h2[i] * di * di + b2[f];
}

// ------------------------------------------------------------------
extern "C" void kernel_launch(void* const* d_in, const int* in_sizes, int n_in,
                              void* d_out, int out_size, void* d_ws, size_t ws_size,
                              hipStream_t stream) {
  const float* x  = (const float*)d_in[0];
  const int*   ei = (const int*)d_in[1];   // [2, E] edge_index
  const float* W1 = (const float*)d_in[2];
  const float* b1 = (const float*)d_in[3];
  const float* W2 = (const float*)d_in[4];
  const float* b2 = (const float*)d_in[5];
  float* out = (float*)d_out;

  const int N = in_sizes[0] / IN_DIMC;
  const int E = in_sizes[1] / 2;
  const int* src = ei;
  const int* dst = ei + E;

  // workspace layout (all fp32, 256B-aligned slices)
  size_t off = 0;
  char* base = (char*)d_ws;
  auto take = [&](size_t count) -> float* {
    float* p = (float*)(base + off);
    off += ((count * sizeof(float) + 255) & ~(size_t)255);
    return p;
  };
  float* dinv = take((size_t)N);              //  0.4 MB  (deg -> dinv in place)
  float* norm = take((size_t)E);              //  6.4 MB
  float* h1   = take((size_t)N * H_DIMC);     // 51.2 MB
  float* agg1 = take((size_t)N * H_DIMC);     // 51.2 MB (becomes relu'd hidden)
  float* h2   = take((size_t)N * OUT_DIMC);   // 16.0 MB
  (void)ws_size; (void)n_in; (void)out_size;

  const int T = 256;
  // degrees & norms
  init_deg   <<<(N + T - 1) / T, T, 0, stream>>>(dinv, N);
  deg_edges  <<<(E + T - 1) / T, T, 0, stream>>>(dst, dinv, E);
  finish_dinv<<<(N + T - 1) / T, T, 0, stream>>>(dinv, N);
  edge_norm  <<<(E + T - 1) / T, T, 0, stream>>>(src, dst, dinv, norm, E);

  // layer 1
  const int nh = N * H_DIMC;
  zero_f32   <<<(nh + T - 1) / T, T, 0, stream>>>(agg1, nh);
  gemm1_wmma <<<(N + 15) / 16, 256, 0, stream>>>(x, W1, h1, N);
  agg_edges_h<<<(E + 7) / 8, 256, 0, stream>>>(src, dst, norm, h1, agg1, E);
  relu_bias1 <<<(nh + T - 1) / T, T, 0, stream>>>(agg1, h1, dinv, b1, nh);

  // layer 2
  const int no = N * OUT_DIMC;
  gemm2_wmma <<<(N + 63) / 64, 384, 0, stream>>>(agg1, W2, h2, N);
  zero_f32   <<<(no + T - 1) / T, T, 0, stream>>>(out, no);
  agg_edges_o<<<(E + 7) / 8, 256, 0, stream>>>(src, dst, norm, h2, out, E);
  final_out  <<<(no + T - 1) / T, T, 0, stream>>>(out, h2, dinv, b2, no);
}